// SimilarityLossWithNegative_6287832121490
// MI455X (gfx1250) — compile-verified
//
#include <hip/hip_runtime.h>
#include <hip/hip_bf16.h>

// Cosine-similarity loss, MI455X (gfx1250, wave32).
// Bandwidth-bound (~17 MB @ 23.3 TB/s). Row-wise dots computed on the matrix
// pipe via chained V_WMMA_F32_16X16X4_F32 (exact f32), diagonal extracted.

typedef __attribute__((ext_vector_type(2))) float v2f;
typedef __attribute__((ext_vector_type(8))) float v8f;

#define DIMD 128
#define TILE 16
#define NEGK 5
#define DP   132          // padded LDS row stride (floats): spreads banks, keeps 8B align
#define CEPS 1e-8f

// One 16x16 cross-dot tile: C[m][n] = sum_k X[m][k] * Y[n][k], accumulated
// over K=128 with 32 chained f32 WMMAs. X,Y are LDS tiles with row stride DP.
// Fragment layout (ISA 7.12.2, 32-bit A 16x4 / B 4x16):
//   lanes 0-15: K-offsets {0,1}; lanes 16-31: K-offsets {2,3}; row = lane&15.
__device__ __forceinline__ v8f dot_chain(const float* X, const float* Y,
                                         int m, int off) {
  v8f c = {};
#pragma unroll
  for (int k0 = 0; k0 < DIMD; k0 += 4) {
    const float2 xa = *(const float2*)&X[m * DP + k0 + off];
    const float2 yb = *(const float2*)&Y[m * DP + k0 + off];
    v2f a; a.x = xa.x; a.y = xa.y;
    v2f b; b.x = yb.x; b.y = yb.y;
    c = __builtin_amdgcn_wmma_f32_16x16x4_f32(
        /*neg_a=*/false, a, /*neg_b=*/false, b,
        /*c_mod=*/(short)0, c, /*reuse_a=*/false, /*reuse_b=*/false);
  }
  return c;
}

// Diagonal of 16x16 f32 C/D: lane l<8 holds d_l in c[l];
// lanes 24..31 hold d_{l-16} in c[l-24]. Store to dst[16] in LDS.
__device__ __forceinline__ void store_diag(const v8f& c, float* dst, int lane) {
  const int idx = (lane < 8) ? lane : (lane - 24);
  float d = 0.0f;
#pragma unroll
  for (int i = 0; i < 8; ++i) d = (idx == i) ? c[i] : d;
  if (lane < 8)        dst[lane]      = d;
  else if (lane >= 24) dst[lane - 16] = d;
}

__global__ __launch_bounds__(32) void cosloss_tile_kernel(
    const float* __restrict__ ue,   // [B,128] user_embeddings
    const float* __restrict__ se,   // [B,128] subreddit_embeddings
    const float* __restrict__ tue,  // [NU,128] total_user_embeddings
    const float* __restrict__ tse,  // [NS,128] total_subreddit_embeddings
    const int*   __restrict__ bu,   // [B] batch_users
    const int*   __restrict__ ni,   // [B,5] neg_indices
    float* __restrict__ partial,    // [2*nTiles] pos/neg partial sums
    int nTiles) {
  __shared__ float lA[TILE * DP];
  __shared__ float lB[TILE * DP];
  __shared__ float dXY[TILE];
  __shared__ float dXX[TILE];
  __shared__ float dYY[TILE];

  const int lane = threadIdx.x;
  const int t    = blockIdx.x;
  const int r0   = t * TILE;
  const int m    = lane & 15;
  const int off  = (lane < 16) ? 0 : 2;
  const int c0   = lane * 4;      // this lane's float4 column chunk

  // ---------------- positive term: cos(user, subreddit) ----------------
#pragma unroll
  for (int j = 0; j < TILE; ++j) {     // coalesced 512B row loads
    *(float4*)&lA[j * DP + c0] = *(const float4*)(ue + (size_t)(r0 + j) * DIMD + c0);
    *(float4*)&lB[j * DP + c0] = *(const float4*)(se + (size_t)(r0 + j) * DIMD + c0);
  }
  __syncthreads();

  v8f cUS = dot_chain(lA, lB, m, off);   // dot(u_i, s_i) on diagonal
  v8f cUU = dot_chain(lA, lA, m, off);   // ||u_i||^2
  v8f cSS = dot_chain(lB, lB, m, off);   // ||s_i||^2
  store_diag(cUS, dXY, lane);
  store_diag(cUU, dXX, lane);
  store_diag(cSS, dYY, lane);
  __syncthreads();

  float posAcc = 0.0f;
  if (lane < TILE) {
    const float nu = fmaxf(sqrtf(dXX[lane]), CEPS);
    const float ns = fmaxf(sqrtf(dYY[lane]), CEPS);
    posAcc = dXY[lane] / (nu * ns);
  }
  __syncthreads();

  // ---------------- negative term: cos(gathered u, 5 sampled negs) ------
#pragma unroll
  for (int j = 0; j < TILE; ++j) {
    const int urow = bu[r0 + j];
    *(float4*)&lA[j * DP + c0] = *(const float4*)(tue + (size_t)urow * DIMD + c0);
  }
  __syncthreads();

  v8f cGG = dot_chain(lA, lA, m, off);   // ||u_b||^2 (gathered)
  store_diag(cGG, dXX, lane);

  float negAcc = 0.0f;
  for (int k = 0; k < NEGK; ++k) {
    __syncthreads();                     // lB free to overwrite
#pragma unroll
    for (int j = 0; j < TILE; ++j) {
      const int srow = ni[(size_t)(r0 + j) * NEGK + k];
      *(float4*)&lB[j * DP + c0] = *(const float4*)(tse + (size_t)srow * DIMD + c0);
    }
    __syncthreads();

    v8f cUN = dot_chain(lA, lB, m, off); // dot(u_b, neg_bk)
    v8f cNN = dot_chain(lB, lB, m, off); // ||neg_bk||^2
    store_diag(cUN, dXY, lane);
    store_diag(cNN, dYY, lane);
    __syncthreads();

    if (lane < TILE) {
      const float nu = fmaxf(sqrtf(dXX[lane]), CEPS);
      const float ns = fmaxf(sqrtf(dYY[lane]), CEPS);
      negAcc += dXY[lane] / (nu * ns);
    }
  }

  // wave32 reduction of the 16 live lanes (others carry 0)
  float p = posAcc, n = negAcc;
#pragma unroll
  for (int s = 16; s > 0; s >>= 1) {
    p += __shfl_down(p, s);
    n += __shfl_down(n, s);
  }
  if (lane == 0) {
    partial[t]          = p;
    partial[nTiles + t] = n;
  }
}

__global__ __launch_bounds__(256) void cosloss_reduce_kernel(
    const float* __restrict__ partial, float* __restrict__ out,
    int nTiles, int Bn) {
  __shared__ float sp[8];
  __shared__ float sn[8];
  float p = 0.0f, n = 0.0f;
  for (int i = threadIdx.x; i < nTiles; i += blockDim.x) {
    p += partial[i];
    n += partial[nTiles + i];
  }
#pragma unroll
  for (int s = 16; s > 0; s >>= 1) {
    p += __shfl_down(p, s);
    n += __shfl_down(n, s);
  }
  const int wid = threadIdx.x >> 5;
  if ((threadIdx.x & 31) == 0) { sp[wid] = p; sn[wid] = n; }
  __syncthreads();
  if (threadIdx.x == 0) {
    float ps = 0.0f, ns = 0.0f;
#pragma unroll
    for (int w = 0; w < 8; ++w) { ps += sp[w]; ns += sn[w]; }
    // loss = (1 - mean(pos)) + mean(neg)
    out[0] = 1.0f - ps / (float)Bn + ns / ((float)Bn * (float)NEGK);
  }
}

extern "C" void kernel_launch(void* const* d_in, const int* in_sizes, int n_in,
                              void* d_out, int out_size, void* d_ws, size_t ws_size,
                              hipStream_t stream) {
  const float* ue  = (const float*)d_in[0];
  const float* se  = (const float*)d_in[1];
  const float* tue = (const float*)d_in[2];
  const float* tse = (const float*)d_in[3];
  const int*   bu  = (const int*)d_in[4];
  const int*   ni  = (const int*)d_in[5];
  float* out = (float*)d_out;
  float* partial = (float*)d_ws;   // 2*nTiles floats, fully written each call

  const int Bn = in_sizes[0] / DIMD;   // 4096
  const int nTiles = Bn / TILE;        // 256

  cosloss_tile_kernel<<<nTiles, 32, 0, stream>>>(ue, se, tue, tse, bu, ni,
                                                 partial, nTiles);
  cosloss_reduce_kernel<<<1, 256, 0, stream>>>(partial, out, nTiles, Bn);
}